// MemoryEfficientAttention_82240033784127
// MI455X (gfx1250) — compile-verified
//
#include <hip/hip_runtime.h>

// Fused top-k (k=32) sparse attention for B=8, L=2048, D=64 (fp32).
// Score GEMM via V_WMMA_F32_16X16X4_F32 (full fp32 to preserve fp64-reference
// ranking); exact per-query top-32 maintained in LDS; weighted V gather fused.

typedef __attribute__((ext_vector_type(2))) float v2f;
typedef __attribute__((ext_vector_type(4))) float v4f;
typedef __attribute__((ext_vector_type(8))) float v8f;

#define NB    8
#define LL    2048
#define DD    64
#define KTOP  32
#define QT    16      // queries per block
#define CHUNK 128     // keys per chunk (4 waves x 2 tiles x 16 keys)
#define SSTR  129     // padded LDS row stride (floats) -> conflict-free banks

__global__ __launch_bounds__(128)
void topk_attn_f32_wmma(const float* __restrict__ q,
                        const float* __restrict__ k,
                        const float* __restrict__ v,
                        float* __restrict__ out)
{
    __shared__ float sChunk[QT * SSTR];   // scores: [query][key_in_chunk]
    __shared__ float sTopV[QT * KTOP];    // per-query top values, descending
    __shared__ int   sTopI[QT * KTOP];    // matching key indices

    const int tid  = threadIdx.x;
    const int wave = tid >> 5;
    const int lane = tid & 31;
    const int row  = lane & 15;   // A: key row in tile / B: query column
    const int half = lane >> 4;   // depth-pair selector

    const int blk   = blockIdx.x;
    const int b     = blk >> 7;        // 128 query tiles per batch
    const int qbase = (blk & 127) * QT;

    const float* __restrict__ Qb = q + ((size_t)b * LL + qbase) * DD;
    const float* __restrict__ Kb = k + (size_t)b * LL * DD;
    const float* __restrict__ Vb = v + (size_t)b * LL * DD;

    // init top lists
    for (int i = tid; i < QT * KTOP; i += 128) {
        sTopV[i] = -__builtin_inff();
        sTopI[i] = 0;
    }

    // Load Q fragment once, pre-scaled by 1/sqrt(D) = 0.125 (exact).
    // B-matrix (Qt) per-lane layout: row = lane%16, depth pair = 4c + 2*half.
    v2f qf[16];
    {
        const float* qp = Qb + row * DD + 2 * half;
#pragma unroll
        for (int c = 0; c < 16; ++c) {
            v2f t = *(const v2f*)(qp + 4 * c);
            qf[c] = t * 0.125f;
        }
    }
    __syncthreads();

    for (int kc = 0; kc < LL; kc += CHUNK) {
        // Each wave computes two independent 16(key)x16(query) S^T tiles.
        const int kb0 = kc + wave * 32;
        const float* kp0 = Kb + (size_t)(kb0 + row) * DD + 2 * half;
        const float* kp1 = kp0 + 16 * DD;
        v8f acc0 = {};
        v8f acc1 = {};
#pragma unroll
        for (int c = 0; c < 16; ++c) {
            v2f a0 = *(const v2f*)(kp0 + 4 * c);
            v2f a1 = *(const v2f*)(kp1 + 4 * c);
            acc0 = __builtin_amdgcn_wmma_f32_16x16x4_f32(
                false, a0, false, qf[c], (short)0, acc0, false, false);
            acc1 = __builtin_amdgcn_wmma_f32_16x16x4_f32(
                false, a1, false, qf[c], (short)0, acc1, false, false);
        }

        // Scatter: lane holds query n=row, keys m = 8*half + r (tile-local).
        {
            float* dst = &sChunk[row * SSTR];
            const int kb = wave * 32 + 8 * half;
#pragma unroll
            for (int r = 0; r < 8; ++r) {
                dst[kb + r]      = acc0[r];
                dst[kb + 16 + r] = acc1[r];
            }
        }
        __syncthreads();

        // Exact top-32 update: one thread per query (spread across waves).
        if ((tid & 7) == 0) {
            const int qq = tid >> 3;
            float* tv = &sTopV[qq * KTOP];
            int*   ti = &sTopI[qq * KTOP];
            const float* sc = &sChunk[qq * SSTR];
            float minv = tv[KTOP - 1];
            for (int j = 0; j < CHUNK; ++j) {
                float s = sc[j];
                if (s > minv) {   // strict: stable, lowest index wins on ties
                    int p = KTOP - 1;
                    while (p > 0 && tv[p - 1] < s) {
                        tv[p] = tv[p - 1];
                        ti[p] = ti[p - 1];
                        --p;
                    }
                    tv[p] = s;
                    ti[p] = kc + j;
                    minv = tv[KTOP - 1];
                }
            }
        }
        __syncthreads();
    }

    // out[b, qbase+qq, :] = sum_j topV[j] * V[topI[j], :]
    // 8 threads per query, 8 dims each; descending-j order matches reference.
    {
        const int qq   = tid >> 3;
        const int dseg = (tid & 7) * 8;
        const float* tvp = &sTopV[qq * KTOP];
        const int*   tip = &sTopI[qq * KTOP];
        float acc[8];
#pragma unroll
        for (int i = 0; i < 8; ++i) acc[i] = 0.0f;
        for (int j = 0; j < KTOP; ++j) {
            const float w = tvp[j];
            const float* vr = Vb + (size_t)tip[j] * DD + dseg;
            v4f a = *(const v4f*)(vr);
            v4f c = *(const v4f*)(vr + 4);
#pragma unroll
            for (int i = 0; i < 4; ++i) {
                acc[i]     += w * a[i];
                acc[i + 4] += w * c[i];
            }
        }
        float* op = out + ((size_t)b * LL + qbase + qq) * DD + dseg;
#pragma unroll
        for (int i = 0; i < 8; ++i) op[i] = acc[i];
    }
}

extern "C" void kernel_launch(void* const* d_in, const int* in_sizes, int n_in,
                              void* d_out, int out_size, void* d_ws, size_t ws_size,
                              hipStream_t stream) {
    (void)in_sizes; (void)n_in; (void)d_ws; (void)ws_size; (void)out_size;
    const float* q = (const float*)d_in[0];
    const float* k = (const float*)d_in[1];
    const float* v = (const float*)d_in[2];
    float* out = (float*)d_out;
    dim3 grid(NB * (LL / QT));   // 1024 blocks: one per (batch, 16-query tile)
    dim3 block(128);             // 4 waves
    topk_attn_f32_wmma<<<grid, block, 0, stream>>>(q, k, v, out);
}